// PwGaANLayer_54228257080050
// MI455X (gfx1250) — compile-verified
//
#include <hip/hip_runtime.h>
#include <math.h>

typedef __attribute__((ext_vector_type(2))) float v2f;
typedef __attribute__((ext_vector_type(8))) float v8f;

#define N_NODES 65536
#define N_EDGES 1048576
#define DIM 64
#define NEG_SLOPE 0.01f
#define IMIN ((int)0x80000000)

// Monotonic int encoding of IEEE float: int compare == float compare.
static __device__ __forceinline__ int f2mono(float f) {
    int i = __float_as_int(f);
    return (i >= 0) ? i : (i ^ 0x7fffffff);
}
static __device__ __forceinline__ float mono2f(int i) {
    return __int_as_float((i >= 0) ? i : (i ^ 0x7fffffff));
}

// ---------------------------------------------------------------- init
__global__ void k_init(float* __restrict__ h, float* __restrict__ msum,
                       int* __restrict__ maxf, float* __restrict__ den,
                       float* __restrict__ deg, int* __restrict__ m) {
    int i = blockIdx.x * blockDim.x + threadIdx.x;   // exactly N*64 threads
    h[i] = 0.f;
    msum[i] = 0.f;
    maxf[i] = IMIN;
    if (i < N_NODES) { den[i] = 0.f; deg[i] = 0.f; m[i] = IMIN; }
}

// ------------------------------------------------- tiny aux weight build
// auxW is a 16x64 "weight" block; row n = column n of the aux B tile.
//   row0 = wl = attl @ Wa, row1 = wr = attr @ Wa, row2 = gl_w, rest 0.
__global__ void k_auxw(const float* __restrict__ Wa, const float* __restrict__ attl,
                       const float* __restrict__ attr, const float* __restrict__ gl,
                       float* __restrict__ auxW) {
    int t = threadIdx.x;                              // 256 threads, 1 block
    for (int i = t; i < 16 * DIM; i += 256) auxW[i] = 0.f;
    __syncthreads();
    if (t < DIM) {
        float wl = 0.f, wr = 0.f;
        for (int j = 0; j < DIM; ++j) {
            float w = Wa[j * DIM + t];
            wl += attl[j] * w;
            wr += attr[j] * w;
        }
        auxW[0 * DIM + t] = wl;
        auxW[1 * DIM + t] = wr;
        auxW[2 * DIM + t] = gl[t];
    }
}

// ---------------------------------------------------- WMMA node GEMM
// zg = v @ Wg^T  (16x64 tile per wave, fp32 WMMA 16x16x4)
// plus aux tile: cols {0,1,2} of  v @ auxW^T  -> zl, zr, glv
__global__ void __launch_bounds__(256) k_gemm(
        const float* __restrict__ v, const float* __restrict__ Wg,
        const float* __restrict__ auxW, float* __restrict__ zg,
        float* __restrict__ zl, float* __restrict__ zr, float* __restrict__ glv) {
    const int lane = threadIdx.x & 31;
    const int wave = threadIdx.x >> 5;
    const int r0 = (blockIdx.x * 8 + wave) * 16;      // 16 rows of v per wave
    const int mm = lane & 15;                          // M (A) / N (B) index
    const int kh = (lane >> 4) << 1;                   // K half-select: 0 or 2

    v8f c0 = {}, c1 = {}, c2 = {}, c3 = {}, ca = {};

    const float* ap  = v    + (size_t)(r0 + mm) * DIM + kh;
    const float* b0p = Wg   + (size_t)(0  + mm) * DIM + kh;
    const float* b1p = Wg   + (size_t)(16 + mm) * DIM + kh;
    const float* b2p = Wg   + (size_t)(32 + mm) * DIM + kh;
    const float* b3p = Wg   + (size_t)(48 + mm) * DIM + kh;
    const float* bap = auxW + (size_t)mm        * DIM + kh;

#pragma unroll
    for (int k = 0; k < DIM; k += 4) {
        v2f a  = *(const v2f*)(ap  + k);
        v2f b0 = *(const v2f*)(b0p + k);
        v2f b1 = *(const v2f*)(b1p + k);
        v2f b2 = *(const v2f*)(b2p + k);
        v2f b3 = *(const v2f*)(b3p + k);
        v2f ba = *(const v2f*)(bap + k);
        c0 = __builtin_amdgcn_wmma_f32_16x16x4_f32(false, a, false, b0, (short)0, c0, false, false);
        c1 = __builtin_amdgcn_wmma_f32_16x16x4_f32(false, a, false, b1, (short)0, c1, false, false);
        c2 = __builtin_amdgcn_wmma_f32_16x16x4_f32(false, a, false, b2, (short)0, c2, false, false);
        c3 = __builtin_amdgcn_wmma_f32_16x16x4_f32(false, a, false, b3, (short)0, c3, false, false);
        ca = __builtin_amdgcn_wmma_f32_16x16x4_f32(false, a, false, ba, (short)0, ca, false, false);
    }

    // C layout: vgpr i -> M = i + 8*(lane>=16), N = lane&15
    const int rbase = r0 + ((lane >> 4) << 3);
    const int col = lane & 15;
#pragma unroll
    for (int i = 0; i < 8; ++i) {
        size_t row = (size_t)(rbase + i) * DIM;
        zg[row + 0  + col] = c0[i];
        zg[row + 16 + col] = c1[i];
        zg[row + 32 + col] = c2[i];
        zg[row + 48 + col] = c3[i];
    }
    if (col < 3) {
        float* dp = (col == 0) ? zl : ((col == 1) ? zr : glv);
#pragma unroll
        for (int i = 0; i < 8; ++i) dp[rbase + i] = ca[i];
    }
}

// ------------------------------------------------ edge pass 1: scores + segmax + deg
__global__ void k_edge1(const float* __restrict__ pre_w, const int* __restrict__ src,
                        const int* __restrict__ dst, const float* __restrict__ zl,
                        const float* __restrict__ zr, float* __restrict__ ebuf,
                        int* __restrict__ m, float* __restrict__ deg) {
    int e = blockIdx.x * blockDim.x + threadIdx.x;
    if (e >= N_EDGES) return;
    float pw = pre_w[e];
    int s = src[e], d = dst[e];
    float a = pw * zl[s] + zr[d];
    float ee = (a >= 0.f) ? a : NEG_SLOPE * a;       // leaky_relu
    ebuf[e] = ee;
    atomicMax(m + d, f2mono(ee));
    atomicAdd(deg + d, 1.0f);
}

// ------------------------------------------------ edge pass 2: exp + segsum den
__global__ void k_edge2(const int* __restrict__ dst, const int* __restrict__ m,
                        float* __restrict__ ebuf, float* __restrict__ den) {
    int e = blockIdx.x * blockDim.x + threadIdx.x;
    if (e >= N_EDGES) return;
    int d = dst[e];
    float ex = __expf(ebuf[e] - mono2f(m[d]));
    ebuf[e] = ex;                                    // reuse buffer for ex
    atomicAdd(den + d, ex);
}

// ------------------------------------------------ edge pass 3: heavy aggregation
// One wave per edge, 2 feature dims per lane -> 256B coalesced rows.
__global__ void __launch_bounds__(256) k_edge3(
        const float* __restrict__ pre_w, const int* __restrict__ src,
        const int* __restrict__ dst, const float* __restrict__ exbuf,
        const float* __restrict__ den, const float* __restrict__ proj_z,
        const float* __restrict__ v, const float* __restrict__ zg,
        float* __restrict__ h, float* __restrict__ msum, int* __restrict__ maxf) {
    int g = blockIdx.x * blockDim.x + threadIdx.x;
    int e = g >> 5;
    int lane = g & 31;
    int s = src[e], d = dst[e];
    float pw = pre_w[e];
    float alpha = exbuf[e] / den[d];
    size_t sb = (size_t)s * DIM + 2 * lane;
    size_t db = (size_t)d * DIM + 2 * lane;

    v2f pz = *(const v2f*)(proj_z + sb);
    atomicAdd(h + db,     alpha * pz[0]);
    atomicAdd(h + db + 1, alpha * pz[1]);

    v2f vv = *(const v2f*)(v + sb);
    atomicAdd(msum + db,     pw * vv[0]);
    atomicAdd(msum + db + 1, pw * vv[1]);

    v2f zz = *(const v2f*)(zg + sb);
    atomicMax(maxf + db,     f2mono(pw * zz[0]));
    atomicMax(maxf + db + 1, f2mono(pw * zz[1]));
}

// ------------------------------------------------ node pass: gate + output
__global__ void __launch_bounds__(256) k_node(
        const float* __restrict__ proj_z, const float* __restrict__ h,
        const float* __restrict__ msum, const int* __restrict__ maxf,
        const float* __restrict__ deg, const float* __restrict__ glv,
        const float* __restrict__ gm_w, const float* __restrict__ gr_w,
        float* __restrict__ out) {
    int g = blockIdx.x * blockDim.x + threadIdx.x;
    int n = g >> 5;
    int lane = g & 31;
    int d0 = 2 * lane;
    size_t nb = (size_t)n * DIM + d0;

    float dn = deg[n];
    float inv = 1.0f / fmaxf(dn, 1.0f);
    float mf0 = (dn > 0.f) ? mono2f(maxf[nb])     : 0.f;
    float mf1 = (dn > 0.f) ? mono2f(maxf[nb + 1]) : 0.f;

    float acc = mf0 * gm_w[d0] + mf1 * gm_w[d0 + 1]
              + msum[nb] * inv * gr_w[d0] + msum[nb + 1] * inv * gr_w[d0 + 1];
#pragma unroll
    for (int off = 16; off > 0; off >>= 1) acc += __shfl_xor(acc, off, 32);

    float gv = 1.0f / (1.0f + __expf(-(glv[n] + acc)));
    out[nb]     = proj_z[nb]     + gv * h[nb];
    out[nb + 1] = proj_z[nb + 1] + gv * h[nb + 1];
}

extern "C" void kernel_launch(void* const* d_in, const int* in_sizes, int n_in,
                              void* d_out, int out_size, void* d_ws, size_t ws_size,
                              hipStream_t stream) {
    const float* v      = (const float*)d_in[0];
    const float* proj_z = (const float*)d_in[1];
    const float* pre_w  = (const float*)d_in[2];
    const float* Wa_w   = (const float*)d_in[3];
    const float* attl_w = (const float*)d_in[4];
    const float* attr_w = (const float*)d_in[5];
    const float* Wg_w   = (const float*)d_in[6];
    const float* gl_w   = (const float*)d_in[7];
    const float* gm_w   = (const float*)d_in[8];
    const float* gr_w   = (const float*)d_in[9];
    const int*   src    = (const int*)d_in[10];
    const int*   dst    = (const int*)d_in[11];
    float* out = (float*)d_out;

    char* ws = (char*)d_ws;
    size_t off = 0;
    auto carve = [&](size_t bytes) -> void* {
        void* p = ws + off;
        off += (bytes + 255) & ~(size_t)255;
        return p;
    };
    float* zg   = (float*)carve((size_t)N_NODES * DIM * 4);
    float* h    = (float*)carve((size_t)N_NODES * DIM * 4);
    float* msum = (float*)carve((size_t)N_NODES * DIM * 4);
    int*   maxf = (int*)  carve((size_t)N_NODES * DIM * 4);
    float* ebuf = (float*)carve((size_t)N_EDGES * 4);
    float* zl   = (float*)carve((size_t)N_NODES * 4);
    float* zr   = (float*)carve((size_t)N_NODES * 4);
    float* glv  = (float*)carve((size_t)N_NODES * 4);
    float* den  = (float*)carve((size_t)N_NODES * 4);
    float* deg  = (float*)carve((size_t)N_NODES * 4);
    int*   m    = (int*)  carve((size_t)N_NODES * 4);
    float* auxW = (float*)carve((size_t)16 * DIM * 4);

    k_init <<<(N_NODES * DIM) / 256, 256, 0, stream>>>(h, msum, maxf, den, deg, m);
    k_auxw <<<1, 256, 0, stream>>>(Wa_w, attl_w, attr_w, gl_w, auxW);
    k_gemm <<<N_NODES / 128, 256, 0, stream>>>(v, Wg_w, auxW, zg, zl, zr, glv);
    k_edge1<<<N_EDGES / 256, 256, 0, stream>>>(pre_w, src, dst, zl, zr, ebuf, m, deg);
    k_edge2<<<N_EDGES / 256, 256, 0, stream>>>(dst, m, ebuf, den);
    k_edge3<<<(N_EDGES * 32) / 256, 256, 0, stream>>>(pre_w, src, dst, ebuf, den,
                                                      proj_z, v, zg, h, msum, maxf);
    k_node <<<(N_NODES * 32) / 256, 256, 0, stream>>>(proj_z, h, msum, maxf, deg,
                                                      glv, gm_w, gr_w, out);
}